// AE_7095285973648
// MI455X (gfx1250) — compile-verified
//
#include <hip/hip_runtime.h>
#include <hip/hip_bf16.h>

// ---------------------------------------------------------------------------
// GCN autoencoder: 4x (WMMA GEMM -> self-loop + edge scatter -> BN(+ReLU))
// Everything is L2-resident on MI455X (192MB L2), so plain global-memory
// kernels with float atomics are the right data-movement strategy.
// GEMMs use V_WMMA_F32_16X16X4_F32 (full fp32 matrix path).
// ---------------------------------------------------------------------------

typedef __attribute__((ext_vector_type(2))) float v2f;
typedef __attribute__((ext_vector_type(8))) float v8f;

#define EPS_BN 1e-5f

// ---------------- degree / normalization ----------------

__global__ void k_zero_f32(float* __restrict__ p, int n) {
    int i = blockIdx.x * blockDim.x + threadIdx.x;
    if (i < n) p[i] = 0.0f;
}

__global__ void k_degree(const long long* __restrict__ dst, float* __restrict__ deg, int e) {
    int i = blockIdx.x * blockDim.x + threadIdx.x;
    if (i < e) atomicAdd(&deg[(int)dst[i]], 1.0f);
}

__global__ void k_dinv(float* __restrict__ deg, int n) {
    int i = blockIdx.x * blockDim.x + threadIdx.x;
    if (i < n) deg[i] = rsqrtf(deg[i] + 1.0f);   // in-degree + self-loop
}

__global__ void k_dsd(const float* __restrict__ dinv, const long long* __restrict__ src,
                      const long long* __restrict__ dst, float* __restrict__ dsd, int e) {
    int i = blockIdx.x * blockDim.x + threadIdx.x;
    if (i < e) dsd[i] = dinv[(int)src[i]] * dinv[(int)dst[i]];
}

// ---------------- WMMA fp32 GEMM:  OUT[M,Nf] = X[M,K] @ W[K,Nf] + bias ------
// One wave per 16-row stripe; 4 waves (64 rows) per block.
// A-fragment layout (32-bit A 16x4): lanes 0-15 hold K=0,1; lanes 16-31 K=2,3.
// C/D layout: VGPR r, lanes 0-15 -> M=r, lanes 16-31 -> M=r+8.

__global__ __launch_bounds__(128) void k_gemm_wmma(
    const float* __restrict__ X, const float* __restrict__ W,
    const float* __restrict__ bias, float* __restrict__ OUT,
    int M, int K, int Nf) {
    const int wave = threadIdx.x >> 5;
    const int lane = threadIdx.x & 31;
    const int half = lane >> 4;       // 0: K pair {0,1} / M base 0 ; 1: {2,3} / +8
    const int m16  = lane & 15;
    const int rowBase = (blockIdx.x * 4 + wave) * 16;
    if (rowBase >= M) return;         // M % 16 == 0 so tiles are always full

    for (int n0 = 0; n0 < Nf; n0 += 16) {
        v8f c = {};
        for (int k0 = 0; k0 < K; k0 += 4) {
            const int ka = k0 + half * 2;
            // A: contiguous float2 per lane (row-major X)
            const float* xp = X + (size_t)(rowBase + m16) * K + ka;
            v2f a; a.x = xp[0]; a.y = xp[1];
            // B: row-major W, two rows of the K-slab, column n0+m16
            v2f b;
            b.x = W[(size_t)(ka + 0) * Nf + n0 + m16];
            b.y = W[(size_t)(ka + 1) * Nf + n0 + m16];
            c = __builtin_amdgcn_wmma_f32_16x16x4_f32(
                    false, a, false, b, (short)0, c, false, false);
        }
        const float bv = bias[n0 + m16];
#pragma unroll
        for (int r = 0; r < 8; ++r) {
            const int row = rowBase + r + half * 8;
            OUT[(size_t)row * Nf + n0 + m16] = c[r] + bv;
        }
    }
}

// ---------------- propagation ----------------

// AGG[i,f] = LIN[i,f] * dinv[i]^2   (self-loop term; also initializes AGG)
__global__ void k_selfloop(const float* __restrict__ LIN, const float* __restrict__ dinv,
                           float* __restrict__ AGG, int lgF, long long total) {
    long long idx = (long long)blockIdx.x * blockDim.x + threadIdx.x;
    if (idx >= total) return;
    int i = (int)(idx >> lgF);
    float di = dinv[i];
    AGG[idx] = LIN[idx] * di * di;
}

// per (edge, float4-chunk): AGG[dst] += LIN[src] * dsd[e]
__global__ void k_scatter(const float* __restrict__ LIN, const float* __restrict__ dsd,
                          const long long* __restrict__ src, const long long* __restrict__ dst,
                          float* __restrict__ AGG, int lgF4, int F, long long total) {
    long long idx = (long long)blockIdx.x * blockDim.x + threadIdx.x;
    if (idx >= total) return;
    long long e = idx >> lgF4;
    int f4 = (int)(idx & ((1 << lgF4) - 1));
    int s = (int)src[e];
    int d = (int)dst[e];
    float w = dsd[e];
    const float4 v = ((const float4*)(LIN + (size_t)s * F))[f4];
    float* out = AGG + (size_t)d * F + f4 * 4;
    atomicAdd(out + 0, v.x * w);
    atomicAdd(out + 1, v.y * w);
    atomicAdd(out + 2, v.z * w);
    atomicAdd(out + 3, v.w * w);
}

// ---------------- batch-norm ----------------

// one block per feature: sum / sumsq of relu(AGG[:, f])
__global__ __launch_bounds__(256) void k_bn_stats(const float* __restrict__ AGG, int F,
                                                  float* __restrict__ sums,
                                                  float* __restrict__ sumsq, int nrows) {
    const int f = blockIdx.x;
    float s = 0.0f, q = 0.0f;
    for (int i = threadIdx.x; i < nrows; i += 256) {
        float x = AGG[(size_t)i * F + f];
        x = fmaxf(x, 0.0f);               // relu(GCN) precedes BN in every layer
        s += x; q += x * x;
    }
    __shared__ float sh[256], shq[256];
    sh[threadIdx.x] = s; shq[threadIdx.x] = q;
    __syncthreads();
    for (int off = 128; off > 0; off >>= 1) {
        if (threadIdx.x < off) {
            sh[threadIdx.x]  += sh[threadIdx.x + off];
            shq[threadIdx.x] += shq[threadIdx.x + off];
        }
        __syncthreads();
    }
    if (threadIdx.x == 0) { sums[f] = sh[0]; sumsq[f] = shq[0]; }
}

// OUT = g*(relu(AGG)-m)*rsqrt(v+eps)+b, optionally relu'd afterwards
__global__ void k_bn_apply(const float* __restrict__ AGG, const float* __restrict__ sums,
                           const float* __restrict__ sumsq, const float* __restrict__ g,
                           const float* __restrict__ b, float* __restrict__ OUT,
                           int lgF, int relu_after, float invN, long long total) {
    long long idx = (long long)blockIdx.x * blockDim.x + threadIdx.x;
    if (idx >= total) return;
    int f = (int)(idx & ((1 << lgF) - 1));
    float x = fmaxf(AGG[idx], 0.0f);
    float m = sums[f] * invN;
    float var = sumsq[f] * invN - m * m;
    float y = g[f] * (x - m) * rsqrtf(var + EPS_BN) + b[f];
    if (relu_after) y = fmaxf(y, 0.0f);
    OUT[idx] = y;
}

// ---------------------------------------------------------------------------

extern "C" void kernel_launch(void* const* d_in, const int* in_sizes, int n_in,
                              void* d_out, int out_size, void* d_ws, size_t ws_size,
                              hipStream_t stream) {
    const int F0 = 256, F1 = 128, F2 = 64;
    const int N = in_sizes[0] / F0;          // 50000
    const int E = in_sizes[1] / 2;           // 800000

    const float*     x   = (const float*)d_in[0];
    const long long* ei  = (const long long*)d_in[1];   // int64 edge_index [2,E]
    const long long* src = ei;
    const long long* dst = ei + E;
    const float *We1 = (const float*)d_in[2],  *be1 = (const float*)d_in[3];
    const float *g1  = (const float*)d_in[4],  *bb1 = (const float*)d_in[5];
    const float *We2 = (const float*)d_in[6],  *be2 = (const float*)d_in[7];
    const float *g2  = (const float*)d_in[8],  *bb2 = (const float*)d_in[9];
    const float *Wd1 = (const float*)d_in[10], *bd1 = (const float*)d_in[11];
    const float *g3  = (const float*)d_in[12], *bb3 = (const float*)d_in[13];
    const float *Wd2 = (const float*)d_in[14], *bd2 = (const float*)d_in[15];
    const float *g4  = (const float*)d_in[16], *bb4 = (const float*)d_in[17];

    // d_out = [h: N*F0][enc1: N*F1][enc2: N*F2]
    float* out_h    = (float*)d_out;
    float* out_enc1 = out_h + (size_t)N * F0;
    float* out_enc2 = out_enc1 + (size_t)N * F1;

    // workspace layout
    float* dinv = (float*)d_ws;                        // N  (deg -> dinv in place)
    float* dsd  = dinv + N;                            // E
    float* bufA = dsd + E;                             // N*256
    float* bufB = bufA + (size_t)N * 256;              // N*256
    float* sums = bufB + (size_t)N * 256;              // 256
    float* sq   = sums + 256;                          // 256

    const int T = 256;
    auto blocks = [](long long n, int t) { return (unsigned)((n + t - 1) / t); };

    // ---- normalization coefficients ----
    k_zero_f32<<<blocks(N, T), T, 0, stream>>>(dinv, N);
    k_degree  <<<blocks(E, T), T, 0, stream>>>(dst, dinv, E);
    k_dinv    <<<blocks(N, T), T, 0, stream>>>(dinv, N);
    k_dsd     <<<blocks(E, T), T, 0, stream>>>(dinv, src, dst, dsd, E);

    // ---- one GCN+BN layer ----
    auto run_layer = [&](const float* X, const float* W, const float* bias,
                         const float* gg, const float* bb, int K, int F,
                         float* LIN, float* AGG, float* OUT, int relu_after) {
        const int lgF  = __builtin_ctz(F);
        const int lgF4 = __builtin_ctz(F / 4);
        const long long totNF = (long long)N * F;
        const long long totEF = (long long)E * (F / 4);

        k_gemm_wmma<<<blocks((long long)N, 64), 128, 0, stream>>>(X, W, bias, LIN, N, K, F);
        k_selfloop <<<blocks(totNF, T), T, 0, stream>>>(LIN, dinv, AGG, lgF, totNF);
        k_scatter  <<<blocks(totEF, T), T, 0, stream>>>(LIN, dsd, src, dst, AGG, lgF4, F, totEF);
        k_bn_stats <<<F, 256, 0, stream>>>(AGG, F, sums, sq, N);
        k_bn_apply <<<blocks(totNF, T), T, 0, stream>>>(AGG, sums, sq, gg, bb, OUT,
                                                        lgF, relu_after, 1.0f / N, totNF);
    };

    // encoder
    run_layer(x,        We1, be1, g1, bb1, F0, F1, bufA, bufB, out_enc1, 1);
    run_layer(out_enc1, We2, be2, g2, bb2, F1, F2, bufA, bufB, out_enc2, 0);
    // decoder (layer 3 output parks in bufA; layer 4 ping-pongs back)
    run_layer(out_enc2, Wd1, bd1, g3, bb3, F2, F1, bufA, bufB, bufA, 1);
    run_layer(bufA,     Wd2, bd2, g4, bb4, F1, F0, bufB, bufA, out_h, 0);
}